// Self_Attention_37108517437756
// MI455X (gfx1250) — compile-verified
//
#include <hip/hip_runtime.h>
#include <hip/hip_bf16.h>

// Self-attention forward, fp32, via CDNA5 V_WMMA_F32_16X16X4_F32 with
// double-buffered GLOBAL_LOAD_ASYNC_TO_LDS_B128 tile staging (ASYNCcnt).
// B=1, S=4096, HIDDEN=768, HEADS=12, HEAD_DIM=64.

#define S_LEN 4096
#define HID   768
#define NH    12
#define HD    64

typedef __attribute__((ext_vector_type(2))) float v2f;
typedef __attribute__((ext_vector_type(8))) float v8f;

// ---- CDNA5 async copy helpers ---------------------------------------------
__device__ __forceinline__ unsigned lds_byte_addr(void* p) {
    // generic -> LDS(AS3) addrspacecast, then ptrtoint => 32-bit LDS offset
    return (unsigned)(unsigned long long)(__attribute__((address_space(3))) void*)p;
}

__device__ __forceinline__ void async_copy_b128(void* lds_dst, const float* gsrc) {
    // GLOBAL_LOAD_ASYNC_TO_LDS_B128, GV mode: per-lane 64-bit global address,
    // per-lane 32-bit LDS destination address. Tracked by ASYNCcnt.
    asm volatile("global_load_async_to_lds_b128 %0, %1, off"
                 :: "v"(lds_byte_addr(lds_dst)),
                    "v"((unsigned long long)(size_t)gsrc)
                 : "memory");
}

template <int N>
__device__ __forceinline__ void wait_asynccnt() {
    asm volatile("s_wait_asynccnt %0" :: "n"(N) : "memory");
}

// ---------------------------------------------------------------------------
// Kernel 1: fused QKV projection.  out[sel] = X @ W[sel] + b[sel]
//   grid = (12 col-tiles, 64 row-tiles, 3 matrices), block = 128 (4 waves).
//   Block computes a 64x64 tile; wave computes 32x32 as 2x2 WMMA C-frags.
//   X(64x32)/W(32x64) tiles double-buffered in LDS via async copy.
// ---------------------------------------------------------------------------
__global__ __launch_bounds__(128)
void qkv_gemm_kernel(const float* __restrict__ X,
                     const float* __restrict__ Wq, const float* __restrict__ bq,
                     const float* __restrict__ Wk, const float* __restrict__ bk,
                     const float* __restrict__ Wv, const float* __restrict__ bv,
                     float* __restrict__ ws) {
    const int sel = blockIdx.z;
    const float* W    = (sel == 0) ? Wq : (sel == 1) ? Wk : Wv;
    const float* bias = (sel == 0) ? bq : (sel == 1) ? bk : bv;
    float* O = ws + (size_t)sel * S_LEN * HID;

    const int row0 = blockIdx.y * 64;
    const int col0 = blockIdx.x * 64;

    const int tid  = threadIdx.x;
    const int wave = tid >> 5;
    const int lane = tid & 31;
    const int m    = lane & 15;         // row (A) / col (B,C) within 16
    const int g    = lane >> 4;         // lane-group
    const int wr   = (wave >> 1) * 32;  // wave row offset in 64x64 tile
    const int wc   = (wave & 1) * 32;   // wave col offset

    __shared__ float Xl[2][64 * 32];    // 2 x 8 KB
    __shared__ float Wl[2][32 * 64];    // 2 x 8 KB

    // Per-thread staging map (512 float4 per tile / 128 threads = 4 each).
    auto issue_tiles = [&](int k0, int buf) {
#pragma unroll
        for (int i = 0; i < 4; ++i) {
            const int idx = tid + i * 128;                 // 0..511
            const int xr = idx >> 3, xc = (idx & 7) << 2;  // X: 64 rows x 8 f4
            async_copy_b128(&Xl[buf][xr * 32 + xc],
                            &X[(size_t)(row0 + xr) * HID + k0 + xc]);
            const int wrw = idx >> 4, wcw = (idx & 15) << 2; // W: 32 x 16 f4
            async_copy_b128(&Wl[buf][wrw * 64 + wcw],
                            &W[(size_t)(k0 + wrw) * HID + col0 + wcw]);
        }
    };

    v8f c[2][2];
#pragma unroll
    for (int i = 0; i < 2; ++i)
#pragma unroll
        for (int j = 0; j < 2; ++j)
            c[i][j] = (v8f){0.f, 0.f, 0.f, 0.f, 0.f, 0.f, 0.f, 0.f};

    constexpr int NKT = HID / 32;  // 24 k-tiles
    issue_tiles(0, 0);             // prologue: tile 0 in flight (8 ops/thread)

    for (int kt = 0; kt < NKT; ++kt) {
        const int cur = kt & 1, nxt = cur ^ 1;
        __syncthreads();           // everyone done reading buffer `nxt`
        if (kt + 1 < NKT) {
            issue_tiles((kt + 1) * 32, nxt);
            wait_asynccnt<8>();    // in-order => tile kt's 8 ops complete
        } else {
            wait_asynccnt<0>();
        }
        __syncthreads();           // all waves' portions of tile kt visible

#pragma unroll
        for (int kb = 0; kb < 8; ++kb) {  // k step of 4 within 32-deep tile
            const int kk = kb * 4 + 2 * g;
            v2f a[2], b[2];
#pragma unroll
            for (int i = 0; i < 2; ++i)
                a[i] = *(const v2f*)&Xl[cur][(wr + i * 16 + m) * 32 + kk];
#pragma unroll
            for (int j = 0; j < 2; ++j) {
                b[j].x = Wl[cur][kk * 64 + wc + j * 16 + m];
                b[j].y = Wl[cur][(kk + 1) * 64 + wc + j * 16 + m];
            }
#pragma unroll
            for (int i = 0; i < 2; ++i)
#pragma unroll
                for (int j = 0; j < 2; ++j)
                    c[i][j] = __builtin_amdgcn_wmma_f32_16x16x4_f32(
                        false, a[i], false, b[j], (short)0, c[i][j], false, false);
        }
    }

    // Epilogue: bias add + store (C layout: row = r + 8*g, col = m).
#pragma unroll
    for (int j = 0; j < 2; ++j) {
        const float bj = bias[col0 + wc + j * 16 + m];
#pragma unroll
        for (int i = 0; i < 2; ++i) {
#pragma unroll
            for (int r = 0; r < 8; ++r) {
                const int orow = row0 + wr + i * 16 + r + 8 * g;
                O[(size_t)orow * HID + col0 + wc + j * 16 + m] = c[i][j][r] + bj;
            }
        }
    }
}

// ---------------------------------------------------------------------------
// Kernel 2: flash-attention, one head per blockIdx.y, 16 query rows per wave.
//   grid = (64 q-groups of 64 rows, 12 heads), block = 128 (4 waves).
//   Online softmax; QK^T and PV via fp32 WMMA. K/V tiles double-buffered in
//   LDS via async copy.
// ---------------------------------------------------------------------------
__global__ __launch_bounds__(128)
void attn_kernel(const float* __restrict__ ws, float* __restrict__ out) {
    const float* Q = ws;
    const float* K = ws + (size_t)S_LEN * HID;
    const float* V = ws + 2 * (size_t)S_LEN * HID;

    const int head = blockIdx.y;
    const int tid  = threadIdx.x;
    const int wave = tid >> 5;
    const int lane = tid & 31;
    const int m    = lane & 15;
    const int g    = lane >> 4;
    const int q0   = blockIdx.x * 64 + wave * 16;  // this wave's 16 query rows
    const int c0   = head * HD;                    // head's column slice

    __shared__ float Kl[2][16 * 64];     // 2 x 4 KB
    __shared__ float Vl[2][16 * 64];     // 2 x 4 KB
    __shared__ float Pl[4][16 * 16];     // per-wave P scratch (C->A relayout)

    // Per-thread staging: 256 float4 per tile / 128 threads = 2 each (x2 K,V).
    auto issue_kv = [&](int kv0, int buf) {
#pragma unroll
        for (int i = 0; i < 2; ++i) {
            const int idx = tid + i * 128;                 // 0..255
            const int r = idx >> 4, c4 = (idx & 15) << 2;  // 16 rows x 16 f4
            async_copy_b128(&Kl[buf][r * 64 + c4],
                            &K[(size_t)(kv0 + r) * HID + c0 + c4]);
            async_copy_b128(&Vl[buf][r * 64 + c4],
                            &V[(size_t)(kv0 + r) * HID + c0 + c4]);
        }
    };

    // Preload Q fragments for the whole 16x64 block (A layout, float2/lane).
    v2f qf[16];
#pragma unroll
    for (int kb = 0; kb < 16; ++kb)
        qf[kb] = *(const v2f*)&Q[(size_t)(q0 + m) * HID + c0 + kb * 4 + 2 * g];

    v8f acc[4];
#pragma unroll
    for (int db = 0; db < 4; ++db)
        acc[db] = (v8f){0.f, 0.f, 0.f, 0.f, 0.f, 0.f, 0.f, 0.f};
    float mrun[8], lrun[8];
#pragma unroll
    for (int r = 0; r < 8; ++r) { mrun[r] = -__builtin_inff(); lrun[r] = 0.f; }

    constexpr int NIT = S_LEN / 16;  // 256 KV tiles
    issue_kv(0, 0);                  // prologue (4 async ops/thread)

    for (int it = 0; it < NIT; ++it) {
        const int cur = it & 1, nxt = cur ^ 1;
        __syncthreads();             // everyone done reading buffer `nxt`
        if (it + 1 < NIT) {
            issue_kv((it + 1) * 16, nxt);
            wait_asynccnt<4>();      // in-order => tile `it`'s 4 ops complete
        } else {
            wait_asynccnt<0>();
        }
        __syncthreads();             // tile `it` visible to all waves

        // Scores S = Q (16x64) . K_blk^T (64x16): 16 WMMAs over k.
        v8f s = (v8f){0.f, 0.f, 0.f, 0.f, 0.f, 0.f, 0.f, 0.f};
#pragma unroll
        for (int kb = 0; kb < 16; ++kb) {
            // B[k][n] = K[kv0+n][c0+k]: contiguous pair within K row -> float2.
            v2f b = *(const v2f*)&Kl[cur][m * 64 + kb * 4 + 2 * g];
            s = __builtin_amdgcn_wmma_f32_16x16x4_f32(
                false, qf[kb], false, b, (short)0, s, false, false);
        }

        // Online softmax. C layout: VGPR r <-> row (r + 8*g); N across lanes.
        float* pl = Pl[wave];
#pragma unroll
        for (int r = 0; r < 8; ++r) {
            float x = s[r] * 0.125f;  // 1/sqrt(64)
            float bmax = x;
            bmax = fmaxf(bmax, __shfl_xor(bmax, 1));
            bmax = fmaxf(bmax, __shfl_xor(bmax, 2));
            bmax = fmaxf(bmax, __shfl_xor(bmax, 4));
            bmax = fmaxf(bmax, __shfl_xor(bmax, 8));
            const float mnew  = fmaxf(mrun[r], bmax);
            const float alpha = __expf(mrun[r] - mnew);
            const float pe    = __expf(x - mnew);
            float bsum = pe;
            bsum += __shfl_xor(bsum, 1);
            bsum += __shfl_xor(bsum, 2);
            bsum += __shfl_xor(bsum, 4);
            bsum += __shfl_xor(bsum, 8);
            lrun[r] = lrun[r] * alpha + bsum;
            mrun[r] = mnew;
#pragma unroll
            for (int db = 0; db < 4; ++db) acc[db][r] *= alpha;
            pl[(r + 8 * g) * 16 + m] = pe;  // C-layout -> row-major in LDS
        }

        // ctx += P (16x16) . V_blk (16x64), per 16-wide d tile. LDS DS ops
        // are in-order within a wave, so the P store/load round-trip is safe.
#pragma unroll
        for (int db = 0; db < 4; ++db) {
#pragma unroll
            for (int kb = 0; kb < 4; ++kb) {
                const int kk = kb * 4 + 2 * g;
                v2f a = *(const v2f*)&pl[m * 16 + kk];  // A layout of P
                v2f b;
                b.x = Vl[cur][kk * 64 + db * 16 + m];   // B[k][n]=V[kv0+k][..+n]
                b.y = Vl[cur][(kk + 1) * 64 + db * 16 + m];
                acc[db] = __builtin_amdgcn_wmma_f32_16x16x4_f32(
                    false, a, false, b, (short)0, acc[db], false, false);
            }
        }
    }

    // Normalize and store: out[S,768], head slice contiguous -> no transpose.
#pragma unroll
    for (int r = 0; r < 8; ++r) {
        const float inv = 1.0f / lrun[r];
        const int orow = q0 + r + 8 * g;
#pragma unroll
        for (int db = 0; db < 4; ++db)
            out[(size_t)orow * HID + c0 + db * 16 + m] = acc[db][r] * inv;
    }
}

// ---------------------------------------------------------------------------
extern "C" void kernel_launch(void* const* d_in, const int* in_sizes, int n_in,
                              void* d_out, int out_size, void* d_ws, size_t ws_size,
                              hipStream_t stream) {
    const float* X  = (const float*)d_in[0];
    const float* Wq = (const float*)d_in[1];
    const float* bq = (const float*)d_in[2];
    const float* Wk = (const float*)d_in[3];
    const float* bk = (const float*)d_in[4];
    const float* Wv = (const float*)d_in[5];
    const float* bv = (const float*)d_in[6];
    float* out = (float*)d_out;
    float* ws  = (float*)d_ws;  // needs 3*4096*768 floats = 37.75 MB

    dim3 g1(HID / 64, S_LEN / 64, 3);   // (12, 64, 3)
    qkv_gemm_kernel<<<g1, dim3(128), 0, stream>>>(X, Wq, bq, Wk, bk, Wv, bv, ws);

    dim3 g2(S_LEN / 64, NH);            // (64, 12)
    attn_kernel<<<g2, dim3(128), 0, stream>>>(ws, out);
}